// GNN_21277267984741
// MI455X (gfx1250) — compile-verified
//
#include <hip/hip_runtime.h>

#define NN 200000
#define NE 12800000
#define NTILES (NN / 16)   // 12500, exact

typedef __attribute__((ext_vector_type(2))) float v2f;
typedef __attribute__((ext_vector_type(8))) float v8f;

// ---------------- workspace layout (all f32/u32 words) ----------------
// [0,        NN)        degcnt (u32)
// [NN,       17*NN)     agg1   (NN x 16 f32)
// [17*NN,    19*NN)     agg2   (NN x 2  f32)
// [19*NN,    20*NN)     dinv   (NN f32)
// [20*NN,    36*NN)     hs1    (NN x 16 f32)
// [36*NN,    38*NN)     hs2    (NN x 2  f32)
// zeroed region: first 19*NN words. Total 38*NN*4 = ~30.4 MB.

__global__ void gnn_zero_kernel(float* __restrict__ p, int n) {
  int i = blockIdx.x * blockDim.x + threadIdx.x;
  if (i < n) p[i] = 0.0f;
}

__global__ void gnn_deg_kernel(const int* __restrict__ dst, unsigned* __restrict__ degcnt) {
  int e = blockIdx.x * blockDim.x + threadIdx.x;
  if (e < NE) atomicAdd(&degcnt[dst[e]], 1u);
}

__global__ void gnn_dinv_kernel(const unsigned* __restrict__ degcnt, float* __restrict__ dinv) {
  int i = blockIdx.x * blockDim.x + threadIdx.x;
  if (i < NN) dinv[i] = rsqrtf((float)degcnt[i] + 1.0f);
}

// hs1[i, :] = (x[i, :] @ W1) * dinv[i], via V_WMMA_F32_16X16X4_F32.
// One wave computes a 16-row x 16-col tile. K=2 padded to 4 with zeros
// (lanes 16..31 hold K=2,3 in the 32-bit A/B layouts -> zero them).
__global__ void gnn_gemm1_wmma_kernel(const float* __restrict__ x,
                                      const float* __restrict__ W1,
                                      const float* __restrict__ dinv,
                                      float* __restrict__ hs1) {
  int wave = (blockIdx.x * blockDim.x + threadIdx.x) >> 5;
  int lane = threadIdx.x & 31;
  if (wave >= NTILES) return;           // uniform per wave: EXEC stays all-1s
  int base = wave << 4;
  int m    = lane & 15;
  int r    = base + m;
  bool lo  = lane < 16;

  float x0 = x[r * 2 + 0];
  float x1 = x[r * 2 + 1];
  float w0 = W1[m];        // W1[0][m]
  float w1 = W1[16 + m];   // W1[1][m]

  v2f a, b;
  a.x = lo ? x0 : 0.0f;    // A: VGPR0 = K0 (lanes<16) / K2 (=0)
  a.y = lo ? x1 : 0.0f;    // A: VGPR1 = K1 (lanes<16) / K3 (=0)
  b.x = lo ? w0 : 0.0f;    // B: VGPR0 = row K0, col=lane / row K2 (=0)
  b.y = lo ? w1 : 0.0f;    // B: VGPR1 = row K1, col=lane / row K3 (=0)

  v8f c = {};
  c = __builtin_amdgcn_wmma_f32_16x16x4_f32(
      /*neg_a=*/false, a, /*neg_b=*/false, b,
      /*c_mod=*/(short)0, c, /*reuse_a=*/false, /*reuse_b=*/false);

  int col   = lane & 15;
  int rowHi = (lane >> 4) << 3;         // lanes 16-31 hold M = r+8
#pragma unroll
  for (int rr = 0; rr < 8; ++rr) {
    int row = base + rowHi + rr;
    hs1[row * 16 + col] = c[rr] * dinv[row];
  }
}

// 16 lanes per edge (lane = feature): coalesced 64B gather + 64B atomic scatter.
__global__ void gnn_edge16_kernel(const int* __restrict__ src, const int* __restrict__ dst,
                                  const float* __restrict__ hs1, float* __restrict__ agg1) {
  int t = blockIdx.x * blockDim.x + threadIdx.x;
  if (t >= NE * 16) return;
  int e = t >> 4;
  int f = t & 15;
  int s = src[e];
  int d = dst[e];
  unsafeAtomicAdd(&agg1[d * 16 + f], hs1[s * 16 + f]);
}

// out1 = relu(dinv*(agg1+hs1) + b1); hs2 = (out1 @ W2) * dinv
__global__ void gnn_finalize1_kernel(const float* __restrict__ agg1,
                                     const float* __restrict__ hs1,
                                     const float* __restrict__ dinv,
                                     const float* __restrict__ b1,
                                     const float* __restrict__ W2,
                                     float* __restrict__ hs2) {
  int i = blockIdx.x * blockDim.x + threadIdx.x;
  if (i >= NN) return;
  float di = dinv[i];
  float h0 = 0.0f, h1 = 0.0f;
#pragma unroll
  for (int f = 0; f < 16; ++f) {
    float v = di * (agg1[i * 16 + f] + hs1[i * 16 + f]) + b1[f];
    v = v > 0.0f ? v : 0.0f;            // relu
    h0 += v * W2[f * 2 + 0];
    h1 += v * W2[f * 2 + 1];
  }
  hs2[i * 2 + 0] = h0 * di;
  hs2[i * 2 + 1] = h1 * di;
}

// 2 lanes per edge for the 2-wide second layer.
__global__ void gnn_edge2_kernel(const int* __restrict__ src, const int* __restrict__ dst,
                                 const float* __restrict__ hs2, float* __restrict__ agg2) {
  int t = blockIdx.x * blockDim.x + threadIdx.x;
  if (t >= NE * 2) return;
  int e = t >> 1;
  int f = t & 1;
  int s = src[e];
  int d = dst[e];
  unsafeAtomicAdd(&agg2[d * 2 + f], hs2[s * 2 + f]);
}

__global__ void gnn_finalize2_kernel(const float* __restrict__ agg2,
                                     const float* __restrict__ hs2,
                                     const float* __restrict__ dinv,
                                     const float* __restrict__ b2,
                                     float* __restrict__ out) {
  int t = blockIdx.x * blockDim.x + threadIdx.x;
  if (t >= NN * 2) return;
  int i = t >> 1;
  int c = t & 1;
  out[t] = dinv[i] * (agg2[t] + hs2[t]) + b2[c];
}

extern "C" void kernel_launch(void* const* d_in, const int* in_sizes, int n_in,
                              void* d_out, int out_size, void* d_ws, size_t ws_size,
                              hipStream_t stream) {
  const float* x  = (const float*)d_in[0];
  const float* W1 = (const float*)d_in[1];
  const float* b1 = (const float*)d_in[2];
  const float* W2 = (const float*)d_in[3];
  const float* b2 = (const float*)d_in[4];
  const int* edge = (const int*)d_in[5];
  const int* src  = edge;          // edge_index[0]
  const int* dst  = edge + NE;     // edge_index[1]
  float* out = (float*)d_out;

  float* ws = (float*)d_ws;
  unsigned* degcnt = (unsigned*)ws;                 // NN u32
  float* agg1 = ws + (size_t)1 * NN;                // NN*16
  float* agg2 = ws + (size_t)17 * NN;               // NN*2
  float* dinv = ws + (size_t)19 * NN;               // NN
  float* hs1  = ws + (size_t)20 * NN;               // NN*16
  float* hs2  = ws + (size_t)36 * NN;               // NN*2

  const int B = 256;

  // 1) zero degcnt + agg1 + agg2 (first 19*NN words)
  {
    int n = 19 * NN;
    gnn_zero_kernel<<<(n + B - 1) / B, B, 0, stream>>>(ws, n);
  }
  // 2) degree counts from dst
  gnn_deg_kernel<<<(NE + B - 1) / B, B, 0, stream>>>(dst, degcnt);
  // 3) dinv = rsqrt(deg + 1)
  gnn_dinv_kernel<<<(NN + B - 1) / B, B, 0, stream>>>(degcnt, dinv);
  // 4) hs1 = (x @ W1) * dinv  via WMMA (one wave per 16-row tile)
  {
    int threads = NTILES * 32;
    gnn_gemm1_wmma_kernel<<<(threads + B - 1) / B, B, 0, stream>>>(x, W1, dinv, hs1);
  }
  // 5) layer-1 edge aggregation (16 lanes / edge)
  {
    long long n = (long long)NE * 16;
    gnn_edge16_kernel<<<(int)((n + B - 1) / B), B, 0, stream>>>(src, dst, hs1, agg1);
  }
  // 6) finalize layer 1 + layer-2 GEMM + scale
  gnn_finalize1_kernel<<<(NN + B - 1) / B, B, 0, stream>>>(agg1, hs1, dinv, b1, W2, hs2);
  // 7) layer-2 edge aggregation (2 lanes / edge)
  {
    int n = NE * 2;
    gnn_edge2_kernel<<<(n + B - 1) / B, B, 0, stream>>>(src, dst, hs2, agg2);
  }
  // 8) finalize output
  {
    int n = NN * 2;
    gnn_finalize2_kernel<<<(n + B - 1) / B, B, 0, stream>>>(agg2, hs2, dinv, b2, out);
  }
}